// DGCNN_23656679866590
// MI455X (gfx1250) — compile-verified
//
#include <hip/hip_runtime.h>
#include <hip/hip_bf16.h>
#include <math.h>

// ---------------------------------------------------------------- constants
#define N_NODES 131072      // NG * NPG
#define N_EDGES 2097152     // N * 16
#define HID     128
#define NGRAPH  128
#define NPG     1024
#define KTOP    60
#define DTOT    385         // 3*128 + 1
#define C1CH    16
#define C2CH    32
#define KW2     5
#define POOL2   30          // K/2
#define CONVW   26          // 30 - 5 + 1
#define DENSE   832         // 32*26

typedef float v2f __attribute__((ext_vector_type(2)));
typedef float v8f __attribute__((ext_vector_type(8)));

// ---- CDNA5 async memory->LDS path (ASYNCcnt), with safe fallback ----------
#if defined(__has_builtin)
# if __has_builtin(__builtin_amdgcn_global_load_async_to_lds_b128)
#  define HAVE_ASYNC_LDS 1
# endif
#endif
#ifndef HAVE_ASYNC_LDS
# define HAVE_ASYNC_LDS 0
#endif

#if HAVE_ASYNC_LDS
// exact parameter types per hipcc diagnostic: b128 moves a 16B int vector;
// global source lives in AS1, LDS destination in AS3.
typedef int v4i_vs __attribute__((__vector_size__(4 * sizeof(int))));
typedef __attribute__((address_space(1))) v4i_vs* async_gptr_t;
typedef __attribute__((address_space(3))) v4i_vs* async_lptr_t;
#endif

#if defined(__has_builtin)
# if __has_builtin(__builtin_amdgcn_s_wait_asynccnt)
#  define ASYNC_WAIT0() __builtin_amdgcn_s_wait_asynccnt(0)
# else
#  define ASYNC_WAIT0() asm volatile("s_wait_asynccnt 0x0" ::: "memory")
# endif
#else
# define ASYNC_WAIT0() asm volatile("s_wait_asynccnt 0x0" ::: "memory")
#endif

// ---------------------------------------------------------------- utility
__global__ void k_fill_zero(float* __restrict__ p, int n) {
  int i = blockIdx.x * blockDim.x + threadIdx.x;
  if (i < n) p[i] = 0.f;
}

// degrees via float atomics (counts); then turned into norms in-place
__global__ void k_degrees(const int* __restrict__ src, const int* __restrict__ dst,
                          float* __restrict__ deg_s, float* __restrict__ deg_d) {
  int e = blockIdx.x * blockDim.x + threadIdx.x;
  if (e < N_EDGES) {
    atomicAdd(&deg_s[src[e]], 1.f);
    atomicAdd(&deg_d[dst[e]], 1.f);
  }
}

__global__ void k_norm(float* __restrict__ deg, int n) {
  int i = blockIdx.x * blockDim.x + threadIdx.x;
  if (i < n) deg[i] = rsqrtf(fmaxf(deg[i], 1.f));   // clip(deg,1)^-0.5
}

__global__ void k_embed(const int* __restrict__ z, const float* __restrict__ zemb,
                        float* __restrict__ h) {
  int i = blockIdx.x * blockDim.x + threadIdx.x;
  if (i < N_NODES * HID) {
    int n = i >> 7, c = i & 127;
    h[i] = zemb[z[n] * HID + c];
  }
}

// ---------------------------------------------------------------- WMMA GEMM
// Y[N,128] = (X * norm_s[:,None]) @ W[128,128], f32 WMMA 16x16x4.
// Block = 64 rows x 128 cols, 8 waves. Each wave owns one 16-col tile and
// computes 4 M-tiles against it -> B fragments loaded once feed 4 WMMAs.
// A tile staged via async memory->LDS copy (ASYNCcnt) when available.
__global__ void k_gemm_scaled_wmma(const float* __restrict__ X,
                                   const float* __restrict__ norm_s,
                                   const float* __restrict__ W,
                                   float* __restrict__ Y) {
  __shared__ float As[64][132];            // 33.8KB; 132%64=4 -> conflict-free
  const int row0 = blockIdx.x * 64;
  const int tid  = threadIdx.x;            // 256 threads = 8 waves

#if HAVE_ASYNC_LDS
  // 64x128 f32 tile = 2048 x 16B chunks; 8 async B128 copies per thread.
#pragma unroll
  for (int it = 0; it < 8; ++it) {
    int ci   = tid + 256 * it;
    int elem = ci << 2;                    // 4 floats per chunk
    int r = elem >> 7, c = elem & 127;     // (r*132 + c)*4 is 16B aligned
    __builtin_amdgcn_global_load_async_to_lds_b128(
        (async_gptr_t)(X + (size_t)(row0 + r) * HID + c),
        (async_lptr_t)&As[r][c], 0, 0);
  }
  ASYNC_WAIT0();
  __syncthreads();
#else
  for (int i = tid; i < 64 * 128; i += 256) {
    int r = i >> 7, c = i & 127;
    As[r][c] = X[(size_t)(row0 + r) * HID + c];
  }
  __syncthreads();
#endif

  const int wave = tid >> 5;
  const int lane = tid & 31;
  const int col0 = wave * 16;
  const int half = lane >> 4;              // 0: lanes 0-15, 1: lanes 16-31
  const int l    = lane & 15;

  // each lane's A rows are fixed: row = 16*m + l (both lane halves)
  float ns[4];
#pragma unroll
  for (int m = 0; m < 4; ++m) ns[m] = norm_s[row0 + 16 * m + l];

  v8f acc0 = {}, acc1 = {}, acc2 = {}, acc3 = {};
  for (int k = 0; k < HID; k += 4) {
    // B 4x16 f32: rows striped across lanes, upper lane-half carries K+2
    v2f b;
    b.x = W[(k + 2 * half + 0) * HID + col0 + l];
    b.y = W[(k + 2 * half + 1) * HID + col0 + l];
    // A 16x4 f32 layout: V0 = K(0|2), V1 = K(1|3) per lane half
    v2f a0, a1, a2, a3;
    a0.x = As[ 0 + l][k + 2 * half + 0] * ns[0];
    a0.y = As[ 0 + l][k + 2 * half + 1] * ns[0];
    a1.x = As[16 + l][k + 2 * half + 0] * ns[1];
    a1.y = As[16 + l][k + 2 * half + 1] * ns[1];
    a2.x = As[32 + l][k + 2 * half + 0] * ns[2];
    a2.y = As[32 + l][k + 2 * half + 1] * ns[2];
    a3.x = As[48 + l][k + 2 * half + 0] * ns[3];
    a3.y = As[48 + l][k + 2 * half + 1] * ns[3];
    acc0 = __builtin_amdgcn_wmma_f32_16x16x4_f32(false, a0, false, b, (short)0, acc0, false, false);
    acc1 = __builtin_amdgcn_wmma_f32_16x16x4_f32(false, a1, false, b, (short)0, acc1, false, false);
    acc2 = __builtin_amdgcn_wmma_f32_16x16x4_f32(false, a2, false, b, (short)0, acc2, false, false);
    acc3 = __builtin_amdgcn_wmma_f32_16x16x4_f32(false, a3, false, b, (short)0, acc3, false, false);
  }
  // D layout: VGPR j -> (M=j, N=l) lanes 0-15 ; (M=j+8, N=l) lanes 16-31
#pragma unroll
  for (int j = 0; j < 8; ++j) {
    Y[(size_t)(row0 +  0 + j + 8 * half) * HID + col0 + l] = acc0[j];
    Y[(size_t)(row0 + 16 + j + 8 * half) * HID + col0 + l] = acc1[j];
    Y[(size_t)(row0 + 32 + j + 8 * half) * HID + col0 + l] = acc2[j];
    Y[(size_t)(row0 + 48 + j + 8 * half) * HID + col0 + l] = acc3[j];
  }
}

// ---------------------------------------------------------------- edges
// agg[dst] += hw[src] * ew  (128 channels; thread = edge x 4-channel group)
__global__ void k_edge_scatter128(const float* __restrict__ hw,
                                  const int* __restrict__ src,
                                  const int* __restrict__ dst,
                                  const float* __restrict__ ew,
                                  float* __restrict__ agg) {
  int idx = blockIdx.x * blockDim.x + threadIdx.x;    // E*32 threads
  int e   = idx >> 5;
  int c4  = (idx & 31) << 2;
  if (e < N_EDGES) {
    float w = ew[e];
    const float4 v = *reinterpret_cast<const float4*>(&hw[src[e] * HID + c4]);
    float* a = &agg[dst[e] * HID + c4];
    atomicAdd(a + 0, v.x * w);
    atomicAdd(a + 1, v.y * w);
    atomicAdd(a + 2, v.z * w);
    atomicAdd(a + 3, v.w * w);
  }
}

__global__ void k_tanh128(const float* __restrict__ agg, const float* __restrict__ norm_d,
                          const float* __restrict__ b, float* __restrict__ x) {
  int i = blockIdx.x * blockDim.x + threadIdx.x;
  if (i < N_NODES * HID) {
    int n = i >> 7, c = i & 127;
    x[i] = tanhf(agg[i] * norm_d[n] + b[c]);
  }
}

// layer 3 (128 -> 1)
__global__ void k_gemv_scaled(const float* __restrict__ X, const float* __restrict__ norm_s,
                              const float* __restrict__ W, float* __restrict__ y) {
  int n = blockIdx.x * blockDim.x + threadIdx.x;
  if (n < N_NODES) {
    float s = 0.f;
    for (int k = 0; k < HID; k += 4) {
      float4 x = *reinterpret_cast<const float4*>(&X[n * HID + k]);
      s += x.x * W[k] + x.y * W[k + 1] + x.z * W[k + 2] + x.w * W[k + 3];
    }
    y[n] = s * norm_s[n];
  }
}

__global__ void k_edge_scatter1(const float* __restrict__ hw, const int* __restrict__ src,
                                const int* __restrict__ dst, const float* __restrict__ ew,
                                float* __restrict__ agg) {
  int e = blockIdx.x * blockDim.x + threadIdx.x;
  if (e < N_EDGES) atomicAdd(&agg[dst[e]], hw[src[e]] * ew[e]);
}

__global__ void k_tanh1(const float* __restrict__ agg, const float* __restrict__ norm_d,
                        const float* __restrict__ b, float* __restrict__ x) {
  int n = blockIdx.x * blockDim.x + threadIdx.x;
  if (n < N_NODES) x[n] = tanhf(agg[n] * norm_d[n] + b[0]);
}

// ---------------------------------------------------------------- sort-pool
// last channel of sorted features == per-node max over all 385 channels
__global__ void k_node_max(const float* __restrict__ x0, const float* __restrict__ x1,
                           const float* __restrict__ x2, const float* __restrict__ x3,
                           float* __restrict__ nm) {
  int n = blockIdx.x * blockDim.x + threadIdx.x;
  if (n < N_NODES) {
    float m = x3[n];
    for (int c = 0; c < HID; ++c) {
      m = fmaxf(m, x0[n * HID + c]);
      m = fmaxf(m, x1[n * HID + c]);
      m = fmaxf(m, x2[n * HID + c]);
    }
    nm[n] = m;
  }
}

// per graph: bitonic sort 1024 (val desc, idx asc tie-break), keep top-60 indices
__global__ void k_topk(const float* __restrict__ node_max, int* __restrict__ topk) {
  __shared__ float sv[NPG];
  __shared__ int   si[NPG];
  int g = blockIdx.x, t = threadIdx.x;   // 1024 threads
  sv[t] = node_max[g * NPG + t];
  si[t] = t;
  __syncthreads();
  for (int ksz = 2; ksz <= NPG; ksz <<= 1) {
    for (int j = ksz >> 1; j > 0; j >>= 1) {
      int ixj = t ^ j;
      if (ixj > t) {
        float v1 = sv[t], v2 = sv[ixj];
        int   i1 = si[t], i2 = si[ixj];
        bool ranks_after = (v1 < v2) || (v1 == v2 && i1 > i2);  // for descending order
        bool swap = ((t & ksz) == 0) ? ranks_after : !ranks_after;
        if (swap) { sv[t] = v2; sv[ixj] = v1; si[t] = i2; si[ixj] = i1; }
      }
      __syncthreads();
    }
  }
  if (t < KTOP) topk[g * KTOP + t] = si[t];
}

// gather the 385-vector of each selected node and sort ascending (bitonic-512)
__global__ void k_pool_sort(const int* __restrict__ topk,
                            const float* __restrict__ x0, const float* __restrict__ x1,
                            const float* __restrict__ x2, const float* __restrict__ x3,
                            float* __restrict__ pooled) {
  __shared__ float s[512];
  int p = blockIdx.x;                    // 0 .. NG*KTOP-1
  int g = p / KTOP;
  int node = g * NPG + topk[p];
  int t = threadIdx.x;                   // 256 threads
  for (int i = t; i < 512; i += 256) {
    float v;
    if (i < 128)      v = x0[node * HID + i];
    else if (i < 256) v = x1[node * HID + i - 128];
    else if (i < 384) v = x2[node * HID + i - 256];
    else if (i == 384) v = x3[node];
    else v = __builtin_huge_valf();      // pad -> lands at the end ascending
    s[i] = v;
  }
  __syncthreads();
  for (int ksz = 2; ksz <= 512; ksz <<= 1) {
    for (int j = ksz >> 1; j > 0; j >>= 1) {
      for (int base = t; base < 512; base += 256) {
        int ixj = base ^ j;
        if (ixj > base) {
          float a = s[base], b = s[ixj];
          bool asc = ((base & ksz) == 0);
          if (asc ? (a > b) : (a < b)) { s[base] = b; s[ixj] = a; }
        }
      }
      __syncthreads();
    }
  }
  for (int i = t; i < DTOT; i += 256) pooled[p * DTOT + i] = s[i];
}

// ---------------------------------------------------------------- head
__global__ void k_conv1(const float* __restrict__ pooled, const float* __restrict__ w,
                        const float* __restrict__ b, float* __restrict__ h1) {
  int idx = blockIdx.x * blockDim.x + threadIdx.x;
  if (idx < NGRAPH * C1CH * KTOP) {
    int g = idx / (C1CH * KTOP);
    int r = idx % (C1CH * KTOP);
    int c = r / KTOP, k = r % KTOP;
    const float* row = &pooled[(g * KTOP + k) * DTOT];
    float s = b[c];
    for (int d = 0; d < DTOT; ++d) s += row[d] * w[c * DTOT + d];
    h1[(g * C1CH + c) * KTOP + k] = fmaxf(s, 0.f);
  }
}

// maxpool(2) -> conv2 -> relu -> lin1 -> relu -> lin2, one block (128 thr) per graph
__global__ void k_head(const float* __restrict__ h1,
                       const float* __restrict__ c2w, const float* __restrict__ c2b,
                       const float* __restrict__ l1w, const float* __restrict__ l1b,
                       const float* __restrict__ l2w, const float* __restrict__ l2b,
                       float* __restrict__ out) {
  __shared__ float p2[C1CH * POOL2];   // 480
  __shared__ float flat[DENSE];        // 832
  __shared__ float h3[128];
  __shared__ float red[128];
  int g = blockIdx.x, t = threadIdx.x;
  for (int i = t; i < C1CH * POOL2; i += 128) {
    int ic = i / POOL2, k = i % POOL2;
    float a = h1[(g * C1CH + ic) * KTOP + 2 * k];
    float b = h1[(g * C1CH + ic) * KTOP + 2 * k + 1];
    p2[i] = fmaxf(a, b);
  }
  __syncthreads();
  for (int i = t; i < DENSE; i += 128) {
    int oc = i / CONVW, x = i % CONVW;
    float s = c2b[oc];
    for (int ic = 0; ic < C1CH; ++ic)
      for (int kw = 0; kw < KW2; ++kw)
        s += p2[ic * POOL2 + x + kw] * c2w[(oc * C1CH + ic) * KW2 + kw];
    flat[i] = fmaxf(s, 0.f);
  }
  __syncthreads();
  {
    float s = l1b[t];
    for (int i = 0; i < DENSE; ++i) s += flat[i] * l1w[i * 128 + t];
    h3[t] = fmaxf(s, 0.f);
  }
  __syncthreads();
  red[t] = h3[t] * l2w[t];
  __syncthreads();
  for (int s = 64; s > 0; s >>= 1) {
    if (t < s) red[t] += red[t + s];
    __syncthreads();
  }
  if (t == 0) out[g] = red[0] + l2b[0];
}

// ---------------------------------------------------------------- launch
extern "C" void kernel_launch(void* const* d_in, const int* in_sizes, int n_in,
                              void* d_out, int out_size, void* d_ws, size_t ws_size,
                              hipStream_t stream) {
  (void)in_sizes; (void)n_in; (void)out_size; (void)ws_size;
  const int*   z    = (const int*)d_in[0];
  const int*   src  = (const int*)d_in[1];
  const int*   dst  = (const int*)d_in[2];
  const float* ew   = (const float*)d_in[3];
  const float* zemb = (const float*)d_in[4];
  const float* W[4] = {(const float*)d_in[5], (const float*)d_in[7],
                       (const float*)d_in[9], (const float*)d_in[11]};
  const float* B[4] = {(const float*)d_in[6], (const float*)d_in[8],
                       (const float*)d_in[10], (const float*)d_in[12]};
  const float* c1w = (const float*)d_in[13]; const float* c1b = (const float*)d_in[14];
  const float* c2w = (const float*)d_in[15]; const float* c2b = (const float*)d_in[16];
  const float* l1w = (const float*)d_in[17]; const float* l1b = (const float*)d_in[18];
  const float* l2w = (const float*)d_in[19]; const float* l2b = (const float*)d_in[20];
  float* out = (float*)d_out;

  // workspace carve-up (256B aligned)
  char* ws = (char*)d_ws;
  size_t off = 0;
  auto carve = [&](size_t bytes) -> char* {
    char* p = ws + off;
    off += (bytes + 255) & ~size_t(255);
    return p;
  };
  const size_t NF = (size_t)N_NODES * HID * sizeof(float);
  float* norm_s = (float*)carve(N_NODES * sizeof(float));
  float* norm_d = (float*)carve(N_NODES * sizeof(float));
  float* h0     = (float*)carve(NF);
  float* x0     = (float*)carve(NF);
  float* x1     = (float*)carve(NF);
  float* x2     = (float*)carve(NF);
  float* x3     = (float*)carve(N_NODES * sizeof(float));
  float* hw     = (float*)carve(NF);
  float* agg    = (float*)carve(NF);
  float* nmax   = (float*)carve(N_NODES * sizeof(float));
  int*   topk   = (int*)carve(NGRAPH * KTOP * sizeof(int));
  float* pooled = (float*)carve((size_t)NGRAPH * KTOP * DTOT * sizeof(float));
  float* h1     = (float*)carve((size_t)NGRAPH * C1CH * KTOP * sizeof(float));

  const int BS = 256;
  auto grid = [](long long n, int bs) { return (unsigned)((n + bs - 1) / bs); };

  // degrees -> norms
  k_fill_zero<<<grid(N_NODES, BS), BS, 0, stream>>>(norm_s, N_NODES);
  k_fill_zero<<<grid(N_NODES, BS), BS, 0, stream>>>(norm_d, N_NODES);
  k_degrees<<<grid(N_EDGES, BS), BS, 0, stream>>>(src, dst, norm_s, norm_d);
  k_norm<<<grid(N_NODES, BS), BS, 0, stream>>>(norm_s, N_NODES);
  k_norm<<<grid(N_NODES, BS), BS, 0, stream>>>(norm_d, N_NODES);

  // embedding
  k_embed<<<grid((long long)N_NODES * HID, BS), BS, 0, stream>>>(z, zemb, h0);

  // three 128->128 GraphConv layers (WMMA GEMM + L2-resident scatter)
  float* xin = h0;
  float* xs[3] = {x0, x1, x2};
  for (int layer = 0; layer < 3; ++layer) {
    k_gemm_scaled_wmma<<<N_NODES / 64, 256, 0, stream>>>(xin, norm_s, W[layer], hw);
    k_fill_zero<<<grid((long long)N_NODES * HID, BS), BS, 0, stream>>>(agg, N_NODES * HID);
    k_edge_scatter128<<<grid((long long)N_EDGES * 32, BS), BS, 0, stream>>>(hw, src, dst, ew, agg);
    k_tanh128<<<grid((long long)N_NODES * HID, BS), BS, 0, stream>>>(agg, norm_d, B[layer], xs[layer]);
    xin = xs[layer];
  }

  // 128->1 layer
  k_gemv_scaled<<<grid(N_NODES, BS), BS, 0, stream>>>(x2, norm_s, W[3], hw);
  k_fill_zero<<<grid(N_NODES, BS), BS, 0, stream>>>(agg, N_NODES);
  k_edge_scatter1<<<grid(N_EDGES, BS), BS, 0, stream>>>(hw, src, dst, ew, agg);
  k_tanh1<<<grid(N_NODES, BS), BS, 0, stream>>>(agg, norm_d, B[3], x3);

  // sort-pool: per-node max -> per-graph top-60 -> sort selected feature rows
  k_node_max<<<grid(N_NODES, BS), BS, 0, stream>>>(x0, x1, x2, x3, nmax);
  k_topk<<<NGRAPH, NPG, 0, stream>>>(nmax, topk);
  k_pool_sort<<<NGRAPH * KTOP, 256, 0, stream>>>(topk, x0, x1, x2, x3, pooled);

  // head
  k_conv1<<<grid((long long)NGRAPH * C1CH * KTOP, BS), BS, 0, stream>>>(pooled, c1w, c1b, h1);
  k_head<<<NGRAPH, 128, 0, stream>>>(h1, c2w, c2b, l1w, l1b, l2w, l2b, out);
}